// ContrastiveLanguageLoss_84713934946579
// MI455X (gfx1250) — compile-verified
//
#include <hip/hip_runtime.h>
#include <math.h>

#define N_POINTS    262144
#define FEAT_DIM    512
#define NUM_LABELS  200
#define NUM_NEG     8
#define IGNORE_LBL  255
#define NEG_THRESH  0.85f
#define EPS_F       1e-7f

#define WAVES_PER_BLOCK 8
#define PTS_PER_WAVE    16
#define PTS_PER_BLOCK   (WAVES_PER_BLOCK * PTS_PER_WAVE)   // 128
#define NUM_BLOCKS      (N_POINTS / PTS_PER_BLOCK)         // 2048
#define NTILES          13                                 // ceil(200/16); rows padded to 208
#define NROWS_PAD       (NTILES * 16)                      // 208
#define KC              32                                 // K-chunk per staging round
#define KCP             36                                 // LDS row stride: banks (lm*36+2*half)%64 cover all 64 once

typedef __attribute__((ext_vector_type(2))) float v2f;
typedef __attribute__((ext_vector_type(8))) float v8f;

// ---------------------------------------------------------------- a2 = ||anchor||^2
__global__ void cll_a2_kernel(const float* __restrict__ anchors, float* __restrict__ a2) {
    int i = threadIdx.x;
    if (i < NUM_LABELS) {
        const float* row = anchors + (size_t)i * FEAT_DIM;
        float s = 0.f;
        #pragma unroll 8
        for (int c = 0; c < FEAT_DIM; ++c) { float v = row[c]; s = fmaf(v, v, s); }
        a2[i] = s;
    }
}

// ---------------------------------------------------------------- fused WMMA GEMM + gather + loss
// Features stream from HBM exactly once (K-chunked); all 13 anchor-column
// accumulator tiles stay resident in VGPRs; anchors' K-slices staged in LDS.
__global__ void __launch_bounds__(256)
cll_main_kernel(const float* __restrict__ features,
                const int*   __restrict__ labels,
                const float* __restrict__ anchors,
                const int*   __restrict__ neg_inds,
                const float* __restrict__ a2,
                float*       __restrict__ out,        // [1 + 2N]
                float*       __restrict__ partials)   // [NUM_BLOCKS]
{
    __shared__ float b_lds[NROWS_PAD * KCP];                     // anchors[:, kc:kc+KC] slice  (~30 KB)
    __shared__ float dots_s[WAVES_PER_BLOCK][PTS_PER_WAVE * 16]; // wave-private 16x16 C spill   (8 KB)
    __shared__ float wred[WAVES_PER_BLOCK];

    const int tid  = threadIdx.x;
    const int lane = tid & 31;
    const int wid  = tid >> 5;
    const int half = lane >> 4;      // 0 => K-pair {k,k+1}, 1 => {k+2,k+3}
    const int lm   = lane & 15;      // A: point row M; B: anchor col N
    const int my_pt = blockIdx.x * PTS_PER_BLOCK + wid * PTS_PER_WAVE + lm;

    // gather targets (lanes >=16 duplicate lanes 0..15; harmless)
    int lbl = labels[my_pt];
    int negs[NUM_NEG];
    #pragma unroll
    for (int j = 0; j < NUM_NEG; ++j) negs[j] = neg_inds[(size_t)my_pt * NUM_NEG + j];

    // 13 resident accumulator tiles (104 VGPRs)
    v8f acc[NTILES];
    #pragma unroll
    for (int t = 0; t < NTILES; ++t)
        acc[t] = (v8f){0.f, 0.f, 0.f, 0.f, 0.f, 0.f, 0.f, 0.f};

    float f2part = 0.f;
    const float* myArow = features + (size_t)my_pt * FEAT_DIM + half * 2;

    for (int kc = 0; kc < FEAT_DIM; kc += KC) {
        __syncthreads();   // previous slice fully consumed
        // stage anchors[:, kc:kc+KC) for all 208 padded rows (coalesced: 32 consecutive floats per row)
        for (int idx = tid; idx < NROWS_PAD * KC; idx += 256) {
            int row = idx >> 5;           // /KC
            int c   = idx & (KC - 1);
            b_lds[row * KCP + c] = (row < NUM_LABELS)
                                 ? anchors[(size_t)row * FEAT_DIM + kc + c] : 0.f;
        }
        __syncthreads();

        // A chunk: 8 x v2f, read once from HBM
        v2f areg[KC / 4];
        #pragma unroll
        for (int i = 0; i < KC / 4; ++i)
            areg[i] = *(const v2f*)(myArow + kc + i * 4);

        #pragma unroll
        for (int i = 0; i < KC / 4; ++i) {
            f2part = fmaf(areg[i].x, areg[i].x, f2part);
            f2part = fmaf(areg[i].y, areg[i].y, f2part);
        }

        const float* bb = &b_lds[lm * KCP + half * 2];
        #pragma unroll
        for (int i = 0; i < KC / 4; ++i) {
            #pragma unroll
            for (int t = 0; t < NTILES; ++t) {
                v2f b = *(const v2f*)(bb + t * 16 * KCP + i * 4);
                acc[t] = __builtin_amdgcn_wmma_f32_16x16x4_f32(false, areg[i], false, b,
                                                               (short)0, acc[t], false, false);
            }
        }
    }

    // combine ||f||^2 halves: lane l and l^16 together cover all K
    float f2 = f2part + __shfl_xor(f2part, 16, 32);

    // spill each C tile through LDS scratch and gather needed columns
    float pos_dot = 0.f;
    float neg_dot[NUM_NEG];
    #pragma unroll
    for (int j = 0; j < NUM_NEG; ++j) neg_dot[j] = 0.f;

    float* ds = dots_s[wid];
    #pragma unroll
    for (int t = 0; t < NTILES; ++t) {
        #pragma unroll
        for (int r = 0; r < 8; ++r)                 // C layout: VGPR r -> M = r + 8*half, N = lm
            ds[(r + half * 8) * 16 + lm] = acc[t][r];

        const int base = t * 16;
        if (lane < 16) {                             // wave-private, in-order LDS
            int d = lbl - base;
            if ((unsigned)d < 16u) pos_dot = ds[lm * 16 + d];
            #pragma unroll
            for (int j = 0; j < NUM_NEG; ++j) {
                int dj = negs[j] - base;
                if ((unsigned)dj < 16u) neg_dot[j] = ds[lm * 16 + dj];
            }
        }
    }

    float contrib = 0.f;
    if (lane < 16) {
        const bool valid = (lbl != IGNORE_LBL);
        const int  lsafe = (lbl < NUM_LABELS) ? lbl : 0;

        float pd2 = f2 - 2.f * pos_dot + a2[lsafe];
        float pos_dist = sqrtf(fmaxf(pd2, 0.f) + EPS_F);

        float nsum = 0.f;
        #pragma unroll
        for (int j = 0; j < NUM_NEG; ++j) {
            float nd2 = f2 - 2.f * neg_dot[j] + a2[negs[j]];
            nsum += sqrtf(fmaxf(nd2, 0.f) + EPS_F);
        }
        float neg_mean = nsum * (1.f / NUM_NEG);

        if (!valid) { pos_dist = 0.f; neg_mean = 0.f; }
        float pos_loss = fmaxf(pos_dist, 0.f);              // relu(pos_dist - 0)
        float neg_loss = fmaxf(NEG_THRESH - neg_mean, 0.f); // relu(0.85 - neg_dist)

        out[1 + my_pt]            = pos_loss;
        out[1 + N_POINTS + my_pt] = neg_loss;
        contrib = pos_loss + neg_loss;
    }

    // deterministic block reduction (no atomics)
    #pragma unroll
    for (int off = 16; off > 0; off >>= 1) contrib += __shfl_down(contrib, off, 32);
    if (lane == 0) wred[wid] = contrib;
    __syncthreads();
    if (tid == 0) {
        float t = 0.f;
        #pragma unroll
        for (int w = 0; w < WAVES_PER_BLOCK; ++w) t += wred[w];
        partials[blockIdx.x] = t;
    }
}

// ---------------------------------------------------------------- final scalar: mean(pos)+mean(neg)
__global__ void cll_reduce_kernel(const float* __restrict__ partials, float* __restrict__ out) {
    __shared__ float s[256];
    int tid = threadIdx.x;
    float v = 0.f;
    for (int i = tid; i < NUM_BLOCKS; i += 256) v += partials[i];
    s[tid] = v;
    __syncthreads();
    #pragma unroll
    for (int off = 128; off > 0; off >>= 1) {
        if (tid < off) s[tid] += s[tid + off];
        __syncthreads();
    }
    if (tid == 0) out[0] = s[0] * (1.f / N_POINTS);
}

extern "C" void kernel_launch(void* const* d_in, const int* in_sizes, int n_in,
                              void* d_out, int out_size, void* d_ws, size_t ws_size,
                              hipStream_t stream) {
    const float* features = (const float*)d_in[0];
    const int*   labels   = (const int*)  d_in[1];
    const float* anchors  = (const float*)d_in[2];
    const int*   neg_inds = (const int*)  d_in[3];
    float* out = (float*)d_out;
    float* ws  = (float*)d_ws;

    float* a2       = ws;          // 200 floats
    float* partials = ws + 256;    // NUM_BLOCKS floats

    cll_a2_kernel<<<1, 256, 0, stream>>>(anchors, a2);
    cll_main_kernel<<<NUM_BLOCKS, 256, 0, stream>>>(features, labels, anchors, neg_inds,
                                                    a2, out, partials);
    cll_reduce_kernel<<<1, 256, 0, stream>>>(partials, out);
}